// Mamba_masked_21431886807229
// MI455X (gfx1250) — compile-verified
//
#include <hip/hip_runtime.h>
#include <hip/hip_bf16.h>
#include <math.h>

#define BB    32
#define LL    512
#define DM    1024
#define DI    2048
#define DS    16
#define DTR   64
#define ROWS  32
#define COLSC 32
#define EPSF  1e-5f

typedef float v2f __attribute__((ext_vector_type(2)));
typedef float v8f __attribute__((ext_vector_type(8)));

__device__ __forceinline__ float siluf(float x) { return x / (1.0f + expf(-x)); }

// ---------------------------------------------------------------------------
// 1) in_proj: xz[b,e,l] = sum_d W[e,d] * hid[b,l,d]   (M=e, N=b*L+l, K=d)
//    fp32 WMMA 16x16x4, one wave = 32x32 tile: 2x2 register blocking ->
//    4 WMMAs per 4 fragment loads (each fragment reused twice)
// ---------------------------------------------------------------------------
__global__ void __launch_bounds__(128) in_proj_wmma(
    const float* __restrict__ W, const float* __restrict__ hid,
    float* __restrict__ xr, float* __restrict__ zr)
{
    const int lane = threadIdx.x;
    const int m    = lane & 15;   // A: row M / B: col N
    const int kh   = lane >> 4;   // K half (K=2*kh, 2*kh+1)
    const int epair = blockIdx.y * blockDim.y + threadIdx.y;  // 0..127 (32 e each)
    const int npair = blockIdx.x;                             // 0..511 (32 cols each)
    const int e0 = epair << 5;
    const int n0 = npair << 5;    // global col = b*L + l (32 cols, same batch)
    const int b  = n0 / LL;
    const int l0 = n0 % LL;

    const float* ap0 = W   + (size_t)(e0 + m) * DM + (kh << 1);
    const float* ap1 = W   + (size_t)(e0 + 16 + m) * DM + (kh << 1);
    const float* bp0 = hid + (size_t)(n0 + m) * DM + (kh << 1);
    const float* bp1 = hid + (size_t)(n0 + 16 + m) * DM + (kh << 1);

    v8f acc00 = {}, acc01 = {}, acc10 = {}, acc11 = {};
    for (int k = 0; k < DM; k += 4) {
        v2f a0 = *(const v2f*)(ap0 + k);
        v2f a1 = *(const v2f*)(ap1 + k);
        v2f b0 = *(const v2f*)(bp0 + k);
        v2f b1 = *(const v2f*)(bp1 + k);
        acc00 = __builtin_amdgcn_wmma_f32_16x16x4_f32(false, a0, false, b0,
                                                      (short)0, acc00, false, false);
        acc01 = __builtin_amdgcn_wmma_f32_16x16x4_f32(false, a0, false, b1,
                                                      (short)0, acc01, false, false);
        acc10 = __builtin_amdgcn_wmma_f32_16x16x4_f32(false, a1, false, b0,
                                                      (short)0, acc10, false, false);
        acc11 = __builtin_amdgcn_wmma_f32_16x16x4_f32(false, a1, false, b1,
                                                      (short)0, acc11, false, false);
    }
    // C layout: acc[v] holds (M = v + 8*kh, N = m)
    float* dst;
    if (e0 < DI) dst = xr + ((size_t)b * DI + e0) * LL + l0;
    else         dst = zr + ((size_t)b * DI + (e0 - DI)) * LL + l0;
    float* dst1 = dst + (size_t)16 * LL;
#pragma unroll
    for (int v = 0; v < 8; ++v) {
        const size_t row = (size_t)(v + (kh << 3)) * LL;
        dst [row + m]      = acc00[v];
        dst [row + 16 + m] = acc01[v];
        dst1[row + m]      = acc10[v];
        dst1[row + 16 + m] = acc11[v];
    }
}

// ---------------------------------------------------------------------------
// 2) fused fwd + reversed causal conv (4 taps) + SiLU
// ---------------------------------------------------------------------------
__global__ void conv_silu_kernel(
    const float* __restrict__ xr,
    const float* __restrict__ cw,  const float* __restrict__ cb,
    const float* __restrict__ cwr, const float* __restrict__ cbr,
    float* __restrict__ xc, float* __restrict__ xcr)
{
    const size_t idx = (size_t)blockIdx.x * blockDim.x + threadIdx.x;
    if (idx >= (size_t)BB * DI * LL) return;
    const int l = (int)(idx % LL);
    const int d = (int)((idx / LL) % DI);
    const float* row = xr + (idx - l);   // xr[b][d][0]
    float af = cb[d], ar = cbr[d];
#pragma unroll
    for (int j = 0; j < 4; ++j) {
        const int t = l - 3 + j;
        if (t >= 0) {
            af += cw [d * 4 + j] * row[t];
            ar += cwr[d * 4 + j] * row[LL - 1 - t];   // x_flip[t] = x[L-1-t]
        }
    }
    xc [idx] = siluf(af);
    xcr[idx] = siluf(ar);
}

// ---------------------------------------------------------------------------
// 3) row_means: per-thread 32 LDS bins (stride 33 -> conflict-free)
// ---------------------------------------------------------------------------
#define RM_T 128
__global__ void __launch_bounds__(RM_T) row_means_kernel(
    const float* __restrict__ xc, const float* __restrict__ xcr,
    const int* __restrict__ ids,
    float* __restrict__ out_f, float* __restrict__ out_r)
{
    __shared__ int   rid[LL];
    __shared__ float af[RM_T * (ROWS + 1)];
    __shared__ float ar[RM_T * (ROWS + 1)];
    const int tid = threadIdx.x;
    const int b   = blockIdx.y;
    const int d   = blockIdx.x * RM_T + tid;
    for (int i = tid; i < LL; i += RM_T) rid[i] = ids[b * LL + i] / COLSC;
    float* myf = af + tid * (ROWS + 1);
    float* myr = ar + tid * (ROWS + 1);
#pragma unroll
    for (int r = 0; r < ROWS; ++r) { myf[r] = 0.f; myr[r] = 0.f; }
    __syncthreads();
    const float* pf = xc  + ((size_t)b * DI + d) * LL;
    const float* pr = xcr + ((size_t)b * DI + d) * LL;
    for (int l = 0; l < LL; ++l) {
        const int r = rid[l];
        myf[r] += pf[l];
        myr[r] += pr[l];
    }
    const float inv = 1.0f / COLSC;
#pragma unroll
    for (int r = 0; r < ROWS; ++r) {
        out_f[((size_t)b * DI + d) * ROWS + r] = myf[r] * inv;
        out_r[((size_t)b * DI + d) * ROWS + r] = myr[r] * inv;
    }
}

// ---------------------------------------------------------------------------
// 4) x_proj (96 x 2048) @ x_c (2048 x 32) per batch, both branches
// ---------------------------------------------------------------------------
__global__ void xproj_kernel(
    const float* __restrict__ wf, const float* __restrict__ wr,
    const float* __restrict__ xc, const float* __restrict__ xcr,
    float* __restrict__ dbl, float* __restrict__ dblr)
{
    const int r = threadIdx.x;
    const int e = blockIdx.x * blockDim.y + threadIdx.y;
    const int b = blockIdx.y;
    if (e >= 96) return;
    const float* pf = xc  + (size_t)b * DI * ROWS + r;
    const float* pr = xcr + (size_t)b * DI * ROWS + r;
    float s = 0.f, sr = 0.f;
    for (int d = 0; d < DI; ++d) {
        s  += wf[e * DI + d] * pf[(size_t)d * ROWS];
        sr += wr[e * DI + d] * pr[(size_t)d * ROWS];
    }
    dbl [((size_t)b * 96 + e) * ROWS + r] = s;
    dblr[((size_t)b * 96 + e) * ROWS + r] = sr;
}

// ---------------------------------------------------------------------------
// 5) dt = dt_proj_w (2048 x 64) @ dt_lr (64 x 32), both branches
// ---------------------------------------------------------------------------
__global__ void dt_kernel(
    const float* __restrict__ wf, const float* __restrict__ wr,
    const float* __restrict__ dbl, const float* __restrict__ dblr,
    float* __restrict__ dtf, float* __restrict__ dtr)
{
    const int r = threadIdx.x;
    const int d = blockIdx.x * blockDim.y + threadIdx.y;
    const int b = blockIdx.y;
    const float* pf = dbl  + (size_t)b * 96 * ROWS + r;   // rows 0..63 = dt_lr
    const float* pr = dblr + (size_t)b * 96 * ROWS + r;
    float s = 0.f, sr = 0.f;
#pragma unroll 8
    for (int j = 0; j < DTR; ++j) {
        s  += wf[d * DTR + j] * pf[j * ROWS];
        sr += wr[d * DTR + j] * pr[j * ROWS];
    }
    dtf[((size_t)b * DI + d) * ROWS + r] = s;
    dtr[((size_t)b * DI + d) * ROWS + r] = sr;
}

// ---------------------------------------------------------------------------
// 6) selective scan over the 32 pooled steps, 16 states in registers
// ---------------------------------------------------------------------------
__global__ void __launch_bounds__(256) scan_kernel(
    const float* __restrict__ uf, const float* __restrict__ ur,
    const float* __restrict__ dtf, const float* __restrict__ dtr,
    const float* __restrict__ dbl, const float* __restrict__ dblr,
    const float* __restrict__ biasf, const float* __restrict__ biasr,
    const float* __restrict__ Alogf, const float* __restrict__ Alogr,
    float* __restrict__ of, float* __restrict__ orr)
{
    __shared__ float Bmf[DS * ROWS], Cmf[DS * ROWS], Bmr[DS * ROWS], Cmr[DS * ROWS];
    const int tid = threadIdx.x;
    const int b   = blockIdx.y;
    const int d   = blockIdx.x * 256 + tid;
    const size_t dbase = (size_t)b * 96 * ROWS;
    for (int i = tid; i < DS * ROWS; i += 256) {
        Bmf[i] = dbl [dbase + (size_t)DTR * ROWS + i];
        Cmf[i] = dbl [dbase + (size_t)(DTR + DS) * ROWS + i];
        Bmr[i] = dblr[dbase + (size_t)DTR * ROWS + i];
        Cmr[i] = dblr[dbase + (size_t)(DTR + DS) * ROWS + i];
    }
    __syncthreads();
    float Af[DS], Ar[DS], hf[DS], hr[DS];
#pragma unroll
    for (int n = 0; n < DS; ++n) {
        Af[n] = -expf(Alogf[d * DS + n]);
        Ar[n] = -expf(Alogr[d * DS + n]);
        hf[n] = 0.f; hr[n] = 0.f;
    }
    const float bf = biasf[d], br = biasr[d];
    const size_t base = ((size_t)b * DI + d) * ROWS;
    for (int r = 0; r < ROWS; ++r) {
        const float xf  = dtf[base + r] + bf;
        const float xr  = dtr[base + r] + br;
        const float dlf = (xf > 20.f) ? xf : log1pf(expf(xf));   // softplus
        const float dlr = (xr > 20.f) ? xr : log1pf(expf(xr));
        const float duf = dlf * uf[base + r];
        const float dur = dlr * ur[base + r];
        float yf = 0.f, yr = 0.f;
#pragma unroll
        for (int n = 0; n < DS; ++n) {
            hf[n] = expf(dlf * Af[n]) * hf[n] + duf * Bmf[n * ROWS + r];
            yf   += hf[n] * Cmf[n * ROWS + r];
            hr[n] = expf(dlr * Ar[n]) * hr[n] + dur * Bmr[n * ROWS + r];
            yr   += hr[n] * Cmr[n * ROWS + r];
        }
        of [base + r] = yf;
        orr[base + r] = yr;
    }
}

// ---------------------------------------------------------------------------
// 7) gather + skip + merge flipped reverse branch + LayerNorm + SiLU(z) gate
//    block = one (b,l); writes g[(b*L+l)][d] coalesced
// ---------------------------------------------------------------------------
__global__ void __launch_bounds__(256) combine_ln_gate_kernel(
    const float* __restrict__ xc, const float* __restrict__ xcr,
    const float* __restrict__ of, const float* __restrict__ orr,
    const float* __restrict__ zr, const int* __restrict__ ids,
    const float* __restrict__ dsk, const float* __restrict__ dskr,
    const float* __restrict__ lnw, const float* __restrict__ lnb,
    float* __restrict__ g)
{
    __shared__ float rs[256], rq[256];
    const int tid  = threadIdx.x;
    const int l    = blockIdx.x;
    const int b    = blockIdx.y;
    const int rfwd = ids[b * LL + l] / COLSC;
    const int rrev = ids[b * LL + (LL - 1 - l)] / COLSC;
    float vals[8];
    float s = 0.f, q = 0.f;
#pragma unroll
    for (int i = 0; i < 8; ++i) {
        const int d = tid + (i << 8);
        const size_t bd = (size_t)b * DI + d;
        const float xf  = xc [bd * LL + l];
        const float xrv = xcr[bd * LL + (LL - 1 - l)];
        const float o   = of [bd * ROWS + rfwd] + dsk [d] * xf;
        const float orv = orr[bd * ROWS + rrev] + dskr[d] * xrv;
        const float y = 0.5f * (o + orv);
        vals[i] = y; s += y; q += y * y;
    }
    rs[tid] = s; rq[tid] = q;
    __syncthreads();
    for (int off = 128; off > 0; off >>= 1) {
        if (tid < off) { rs[tid] += rs[tid + off]; rq[tid] += rq[tid + off]; }
        __syncthreads();
    }
    const float mu  = rs[0] / DI;
    const float var = rq[0] / DI - mu * mu;
    const float inv = rsqrtf(var + EPSF);
#pragma unroll
    for (int i = 0; i < 8; ++i) {
        const int d = tid + (i << 8);
        const float yn = (vals[i] - mu) * inv * lnw[d] + lnb[d];
        const float zv = zr[((size_t)b * DI + d) * LL + l];
        g[((size_t)b * LL + l) * DI + d] = yn * siluf(zv);
    }
}

// ---------------------------------------------------------------------------
// 8) out_proj: out[bl,o] = sum_d g[bl,d] * W2[o,d]  (M=bl, N=o -> coalesced C)
//    2x2 register blocking: wave computes 32(bl) x 32(o)
// ---------------------------------------------------------------------------
__global__ void __launch_bounds__(128) out_proj_wmma(
    const float* __restrict__ W2, const float* __restrict__ g,
    float* __restrict__ out)
{
    const int lane = threadIdx.x;
    const int m    = lane & 15;
    const int kh   = lane >> 4;
    const int mpair = blockIdx.y * blockDim.y + threadIdx.y;  // 0..511 (32 bl each)
    const int opair = blockIdx.x;                             // 0..31  (32 o each)
    const int bl0 = mpair << 5;
    const int o0  = opair << 5;

    const float* ap0 = g  + (size_t)(bl0 + m) * DI + (kh << 1);
    const float* ap1 = g  + (size_t)(bl0 + 16 + m) * DI + (kh << 1);
    const float* bp0 = W2 + (size_t)(o0 + m) * DI + (kh << 1);        // B = W2^T
    const float* bp1 = W2 + (size_t)(o0 + 16 + m) * DI + (kh << 1);

    v8f acc00 = {}, acc01 = {}, acc10 = {}, acc11 = {};
    for (int k = 0; k < DI; k += 4) {
        v2f a0 = *(const v2f*)(ap0 + k);
        v2f a1 = *(const v2f*)(ap1 + k);
        v2f b0 = *(const v2f*)(bp0 + k);
        v2f b1 = *(const v2f*)(bp1 + k);
        acc00 = __builtin_amdgcn_wmma_f32_16x16x4_f32(false, a0, false, b0,
                                                      (short)0, acc00, false, false);
        acc01 = __builtin_amdgcn_wmma_f32_16x16x4_f32(false, a0, false, b1,
                                                      (short)0, acc01, false, false);
        acc10 = __builtin_amdgcn_wmma_f32_16x16x4_f32(false, a1, false, b0,
                                                      (short)0, acc10, false, false);
        acc11 = __builtin_amdgcn_wmma_f32_16x16x4_f32(false, a1, false, b1,
                                                      (short)0, acc11, false, false);
    }
    float* dst  = out + (size_t)bl0 * DM + o0;
    float* dst1 = dst + (size_t)16 * DM;
#pragma unroll
    for (int v = 0; v < 8; ++v) {
        const size_t row = (size_t)(v + (kh << 3)) * DM;
        dst [row + m]      = acc00[v];
        dst [row + 16 + m] = acc01[v];
        dst1[row + m]      = acc10[v];
        dst1[row + 16 + m] = acc11[v];
    }
}

// ---------------------------------------------------------------------------
extern "C" void kernel_launch(void* const* d_in, const int* in_sizes, int n_in,
                              void* d_out, int out_size, void* d_ws, size_t ws_size,
                              hipStream_t stream)
{
    (void)in_sizes; (void)n_in; (void)out_size; (void)ws_size;
    const float* hid   = (const float*)d_in[0];
    const int*   ids   = (const int*)  d_in[1];
    const float* inW   = (const float*)d_in[2];
    const float* cw    = (const float*)d_in[3];
    const float* cb    = (const float*)d_in[4];
    const float* cwr   = (const float*)d_in[5];
    const float* cbr   = (const float*)d_in[6];
    const float* xpw   = (const float*)d_in[7];
    const float* xpwr  = (const float*)d_in[8];
    const float* dtw   = (const float*)d_in[9];
    const float* dtb   = (const float*)d_in[10];
    const float* dtwr  = (const float*)d_in[11];
    const float* dtbr  = (const float*)d_in[12];
    const float* Alog  = (const float*)d_in[13];
    const float* Alogr = (const float*)d_in[14];
    const float* Dsk   = (const float*)d_in[15];
    const float* Dskr  = (const float*)d_in[16];
    const float* lnw   = (const float*)d_in[17];
    const float* lnb   = (const float*)d_in[18];
    const float* W2    = (const float*)d_in[19];
    float* out = (float*)d_out;

    float* ws = (float*)d_ws;
    const size_t BDL = (size_t)BB * DI * LL;     // 33,554,432 floats
    const size_t BDR = (size_t)BB * DI * ROWS;   //  2,097,152 floats
    float* x_raw  = ws;                          // reused as g later
    float* z      = x_raw  + BDL;
    float* xconv  = z      + BDL;
    float* xconvr = xconv  + BDL;
    float* xc     = xconvr + BDL;
    float* xcr    = xc     + BDR;
    float* dbl    = xcr    + BDR;
    float* dblr   = dbl    + (size_t)BB * 96 * ROWS;
    float* dtf    = dblr   + (size_t)BB * 96 * ROWS;
    float* dtrv   = dtf    + BDR;
    float* orows  = dtrv   + BDR;
    float* orowsr = orows  + BDR;
    float* g      = x_raw;                       // x_raw dead after conv

    // 4096/32 = 128 e-pairs, 16384/32 = 512 n-pairs
    in_proj_wmma<<<dim3(512, 32), dim3(32, 4), 0, stream>>>(inW, hid, x_raw, z);

    const size_t nconv = BDL;
    conv_silu_kernel<<<(unsigned)((nconv + 255) / 256), 256, 0, stream>>>(
        x_raw, cw, cb, cwr, cbr, xconv, xconvr);

    row_means_kernel<<<dim3(DI / RM_T, BB), RM_T, 0, stream>>>(
        xconv, xconvr, ids, xc, xcr);

    xproj_kernel<<<dim3(12, BB), dim3(32, 8), 0, stream>>>(
        xpw, xpwr, xc, xcr, dbl, dblr);

    dt_kernel<<<dim3(DI / 8, BB), dim3(32, 8), 0, stream>>>(
        dtw, dtwr, dbl, dblr, dtf, dtrv);

    scan_kernel<<<dim3(DI / 256, BB), 256, 0, stream>>>(
        xc, xcr, dtf, dtrv, dbl, dblr, dtb, dtbr, Alog, Alogr, orows, orowsr);

    combine_ln_gate_kernel<<<dim3(LL, BB), 256, 0, stream>>>(
        xconv, xconvr, orows, orowsr, z, ids, Dsk, Dskr, lnw, lnb, g);

    // 16384/32 = 512 bl-pairs, 1024/32 = 32 o-pairs
    out_proj_wmma<<<dim3(32, 128), dim3(32, 4), 0, stream>>>(W2, g, out);
}